// ComNet_25975962206687
// MI455X (gfx1250) — compile-verified
//
#include <hip/hip_runtime.h>
#include <math.h>

// Problem constants (fixed by the reference)
#define R_ 64
#define T_ 256
#define N_ 1024
#define PRE_STRIDE 12   // 10 hidden pre-activations padded to 12 floats (48B, 16B-aligned rows)

typedef float v2f __attribute__((ext_vector_type(2)));
typedef float v8f __attribute__((ext_vector_type(8)));

// ---- gfx1250 tanh (TRANS pipe) ------------------------------------------------
#if defined(__AMDGCN__) && __has_builtin(__builtin_amdgcn_tanhf)
#  define FAST_TANH(x) __builtin_amdgcn_tanhf(x)
#elif defined(__AMDGCN__) && __has_builtin(__builtin_amdgcn_tanh_f32)
#  define FAST_TANH(x) __builtin_amdgcn_tanh_f32(x)
#else
#  define FAST_TANH(x) tanhf(x)
#endif

#if defined(__AMDGCN__) && __has_builtin(__builtin_amdgcn_wmma_f32_16x16x4_f32)
#  define HAVE_WMMA_F32_K4 1
#else
#  define HAVE_WMMA_F32_K4 0
#endif

// Split workgroup barrier (ID -1), same mnemonics the toolchain emits for the
// probe-confirmed cluster barrier (-3). "memory" clobber pins LDS op ordering.
#if defined(__AMDGCN__)
#  define WG_DSCNT0()  asm volatile("s_wait_dscnt 0x0" ::: "memory")
#  define WG_SIGNAL()  asm volatile("s_barrier_signal -1" ::: "memory")
#  define WG_WAIT()    asm volatile("s_barrier_wait -1" ::: "memory")
#else
#  define WG_DSCNT0()
#  define WG_SIGNAL()
#  define WG_WAIT()
#endif

// ============================================================================
// Phase A: pre[cell][n] = b1[n] + W1[n][0]*x0 + W1[n][1]*x1  for all 16.8M cells
// via V_WMMA_F32_16X16X4_F32:  A(16x4 rows [x0,x1,1,0]) x B(4x16 = [w1c0;w1c1;b1;0])
// One wave handles a 16-cell tile per iteration. Bandwidth + matrix-pipe work.
// ============================================================================
__global__ __launch_bounds__(256) void comnet_precompute_wmma(
    const float* __restrict__ runs, const float* __restrict__ w1,
    const float* __restrict__ b1, float* __restrict__ pre) {
  const int lane = threadIdx.x & 31;
  const int half = lane >> 4;   // 0: K=0,1 half; 1: K=2,3 half
  const int n    = lane & 15;   // output column / A row within half
  const int gwave  = blockIdx.x * (blockDim.x >> 5) + (threadIdx.x >> 5);
  const int nwaves = gridDim.x * (blockDim.x >> 5);
  const int NTILES = (R_ * T_ * N_) / 16;

#if HAVE_WMMA_F32_K4
  // B operand (loop-invariant): B[0][n]=w1[n][0], B[1][n]=w1[n][1], B[2][n]=b1[n], B[3][n]=0
  v2f bmat;
  if (half == 0) {
    bmat.x = (n < 10) ? w1[n * 4 + 0] : 0.f;
    bmat.y = (n < 10) ? w1[n * 4 + 1] : 0.f;
  } else {
    bmat.x = (n < 10) ? b1[n] : 0.f;
    bmat.y = 0.f;
  }
  for (int tile = gwave; tile < NTILES; tile += nwaves) {
    v2f amat;
    if (half == 0) {  // lanes 0-15: cell (tile*16+n) sensing pair -> A[n][0],A[n][1]
      const float* xp = runs + ((size_t)tile * 16 + n) * 2;
      amat.x = xp[0];
      amat.y = xp[1];
    } else {          // lanes 16-31: A[n][2]=1 (bias feed), A[n][3]=0
      amat.x = 1.f;
      amat.y = 0.f;
    }
    v8f c = {};
    v8f d = __builtin_amdgcn_wmma_f32_16x16x4_f32(false, amat, false, bmat,
                                                  (short)0, c, false, false);
    if (n < PRE_STRIDE) {  // D row m lives in VGPR m%8, lane-half m/8; col = lane
      float* prow = pre + ((size_t)tile * 16 + half * 8) * PRE_STRIDE + n;
#pragma unroll
      for (int m = 0; m < 8; ++m) prow[(size_t)m * PRE_STRIDE] = d[m];
    }
  }
#else
  // Scalar fallback (also what the host-side pass typechecks)
  for (int tile = gwave; tile < NTILES; tile += nwaves) {
    if (half == 0) {
      const size_t cell = (size_t)tile * 16 + n;
      const float x0 = runs[cell * 2 + 0], x1 = runs[cell * 2 + 1];
      for (int k = 0; k < PRE_STRIDE; ++k)
        pre[cell * PRE_STRIDE + k] =
            (k < 10) ? fmaf(w1[k * 4 + 1], x1, fmaf(w1[k * 4 + 0], x0, b1[k])) : 0.f;
    }
  }
#endif
}

// ============================================================================
// Phase B: wavefront scan. One workgroup per run, thread tid == timestep t.
// Step phi processes cells (t, i=phi-2t); comm lives in LDS; reads at step phi
// touch parity (phi&1) addresses, writes the opposite parity -> one split
// barrier round per step. Control output + store are done post-signal (off the
// cross-thread critical path).
// ============================================================================
template <bool USE_PRE>
__global__ __launch_bounds__(T_) void comnet_scan(
    const float* __restrict__ runs, const float* __restrict__ comm0,
    const float* __restrict__ w1, const float* __restrict__ b1,
    const float* __restrict__ w2, const float* __restrict__ b2,
    const float* __restrict__ pre, float* __restrict__ out) {
  __shared__ float Cm[N_ + 2];
  const int r = blockIdx.x;
  const int t = threadIdx.x;

  for (int j = t; j < N_ + 2; j += T_)
    Cm[j] = (j == 0 || j == N_ + 1) ? 0.f : comm0[(size_t)r * N_ + (j - 1)];

  // Weights (uniform across the wave -> scalar regs)
  float a2[10], a3[10], v0[10], v1[10];
  float a0[10], a1[10], bh[10];
#pragma unroll
  for (int k = 0; k < 10; ++k) {
    a2[k] = w1[k * 4 + 2];  // comm-left column
    a3[k] = w1[k * 4 + 3];  // comm-right column
    v0[k] = w2[k];          // W2 row 0 (control)
    v1[k] = w2[10 + k];     // W2 row 1 (comm update)
    if (!USE_PRE) { a0[k] = w1[k * 4 + 0]; a1[k] = w1[k * 4 + 1]; bh[k] = b1[k]; }
  }
  const float c0 = b2[0], c1 = b2[1];

  const size_t cellbase = ((size_t)r * T_ + t) * (size_t)N_;
  const float* prow = pre + cellbase * PRE_STRIDE;
  const float* xrow = runs + cellbase * 2;
  float* orow = out + cellbase;

  __syncthreads();  // comm init visible to all waves

  const int PHI_MAX = 2 * (T_ - 1) + (N_ - 1);
  for (int phi = 0; phi <= PHI_MAX; ++phi) {
    const int i = phi - 2 * t;
    const bool active = (0 <= i && i < N_);
    float h[10];

    // ---- cross-thread critical path: read comm -> h -> o1 -> write comm ----
    if (active) {
      const float cl = Cm[i];
      const float cr = Cm[i + 2];
      if (USE_PRE) {
        // prow + i*12 is 16B-aligned (48B stride from 16B-aligned base)
        const float4* P = (const float4*)(prow + (size_t)i * PRE_STRIDE);
        const float4 q0 = P[0];
        const float4 q1 = P[1];
        const float2 q2 = ((const float2*)P)[4];
        const float p[10] = {q0.x, q0.y, q0.z, q0.w, q1.x,
                             q1.y, q1.z, q1.w, q2.x, q2.y};
#pragma unroll
        for (int k = 0; k < 10; ++k)
          h[k] = FAST_TANH(fmaf(a3[k], cr, fmaf(a2[k], cl, p[k])));
      } else {
        const float2 xv = *(const float2*)(xrow + 2 * (size_t)i);
#pragma unroll
        for (int k = 0; k < 10; ++k)
          h[k] = FAST_TANH(fmaf(a3[k], cr,
                      fmaf(a2[k], cl, fmaf(a1[k], xv.y, fmaf(a0[k], xv.x, bh[k])))));
      }
      float o1 = c1;
#pragma unroll
      for (int k = 0; k < 10; ++k) o1 = fmaf(v1[k], h[k], o1);
      Cm[i + 1] = o1;  // opposite parity of all step-phi reads: race-free
    }

    WG_DSCNT0();   // comm write committed to LDS
    WG_SIGNAL();   // release the other waves

    // ---- off-path work overlaps other waves' barrier arrival ----
    if (active) {
      float o0 = c0;
#pragma unroll
      for (int k = 0; k < 10; ++k) o0 = fmaf(v0[k], h[k], o0);
      orow[i] = o0;
      if (USE_PRE)
        __builtin_prefetch(prow + (size_t)(i + 16) * PRE_STRIDE, 0, 0);
      else
        __builtin_prefetch(xrow + 2 * (size_t)i + 64, 0, 0);
    }

    WG_WAIT();     // all step-phi comm writes visible before step phi+1 reads
  }
}

// ============================================================================
extern "C" void kernel_launch(void* const* d_in, const int* in_sizes, int n_in,
                              void* d_out, int out_size, void* d_ws, size_t ws_size,
                              hipStream_t stream) {
  const float* runs  = (const float*)d_in[0];
  const float* comm0 = (const float*)d_in[1];
  const float* w1    = (const float*)d_in[2];
  const float* b1    = (const float*)d_in[3];
  const float* w2    = (const float*)d_in[4];
  const float* b2    = (const float*)d_in[5];
  float* out = (float*)d_out;

  const size_t pre_bytes = (size_t)R_ * T_ * N_ * PRE_STRIDE * sizeof(float);
  if (d_ws && ws_size >= pre_bytes) {
    float* pre = (float*)d_ws;
    comnet_precompute_wmma<<<dim3(4096), dim3(256), 0, stream>>>(runs, w1, b1, pre);
    comnet_scan<true><<<dim3(R_), dim3(T_), 0, stream>>>(runs, comm0, w1, b1, w2, b2, pre, out);
  } else {
    comnet_scan<false><<<dim3(R_), dim3(T_), 0, stream>>>(runs, comm0, w1, b1, w2, b2, nullptr, out);
  }
}